// MyConv2D_Cleaner2_7292854468611
// MI455X (gfx1250) — compile-verified
//
#include <hip/hip_runtime.h>

typedef __attribute__((ext_vector_type(2))) float v2f;
typedef __attribute__((ext_vector_type(8))) float v8f;

#define CONV_H 128
#define CONV_W 128
#define CONV_IC 8
#define CONV_OC 16
#define K_STEPS 18   // K = 3*3*8 = 72 = 18 * 4
#define MTILES 4     // 4 x (16x16) output tiles per wave = 64 spatial positions

__global__ __launch_bounds__(256)
void conv2d_nhwc_wmma_f32(const float* __restrict__ x,
                          const float* __restrict__ wgt,
                          float* __restrict__ out) {
  const int lane = threadIdx.x & 31;
  const int wave = threadIdx.x >> 5;
  const int job  = blockIdx.x * 8 + wave;      // 0 .. 8191
  const int wseg = job & 1;                    // 2 segments of 64 along W
  const int h    = (job >> 1) & 127;
  const int n    = job >> 8;

  const int half = lane >> 4;                  // 0: K={0,1}, 1: K={2,3} of each step
  const int ml   = lane & 15;                  // A row-in-tile / B+D column (oc)
  const int w0   = wseg * 64;

  // ---- Preload all 18 B fragments (weights, HWIO: [k][oc], oc fastest);
  //      shared across all 4 M-subtiles. ----
  v2f bfrag[K_STEPS];
#pragma unroll
  for (int kk = 0; kk < K_STEPS; ++kk) {
    const int k0 = 4 * kk + 2 * half;          // this lane's first K of the step
    bfrag[kk].x = wgt[k0 * CONV_OC + ml];
    bfrag[kk].y = wgt[(k0 + 1) * CONV_OC + ml];
  }

  const float* xN = x + (size_t)n * CONV_H * CONV_W * CONV_IC;

  v8f acc[MTILES];
#pragma unroll
  for (int t = 0; t < MTILES; ++t) acc[t] = (v8f){};

#pragma unroll
  for (int kk = 0; kk < K_STEPS; ++kk) {
    // k = ((fh*3)+fw)*8 + ic ; the lane's pair (4kk+2*half, +1) never crosses
    // an ic-octet, so fh/fw depend only on kk and the pair is 2 consecutive
    // channels -> one aligned 8-byte load.
    const int pix = kk >> 1;                   // 0..8
    const int fh  = pix / 3;                   // 0..2
    const int fw  = pix % 3;                   // 0..2
    const int ic  = (kk & 1) * 4 + 2 * half;   // 0,2,4,6 (even -> 8B aligned)

    const int  ih   = h + fh - 1;              // wave-uniform
    const int  ihc  = ih < 0 ? 0 : (ih > CONV_H - 1 ? CONV_H - 1 : ih);
    const bool ihok = (ih >= 0) & (ih < CONV_H);
    const float* rowp = xN + (size_t)ihc * CONV_W * CONV_IC + ic;

#pragma unroll
    for (int t = 0; t < MTILES; ++t) {
      const int iw  = w0 + 16 * t + ml + fw - 1;  // per-lane
      const int iwc = iw < 0 ? 0 : (iw > CONV_W - 1 ? CONV_W - 1 : iw);
      // Branchless SAME padding: clamped (always-legal) load, zero via mask
      // multiply; EXEC stays all-1s as WMMA requires.
      const float msk = (ihok & (iw >= 0) & (iw < CONV_W)) ? 1.0f : 0.0f;

      v2f a = *(const v2f*)(rowp + (size_t)iwc * CONV_IC);
      a.x *= msk;
      a.y *= msk;

      // D = A(16x4) * B(4x16) + C ; 4 independent accumulators give RAW
      // distance 4 between WMMAs on the same tile (hides XDL latency).
      acc[t] = __builtin_amdgcn_wmma_f32_16x16x4_f32(
          /*neg_a=*/false, a, /*neg_b=*/false, bfrag[kk],
          /*c_mod=*/(short)0, acc[t], /*reuse_a=*/false, /*reuse_b=*/false);
    }
  }

  // ---- Store D: per subtile, VGPR j -> row (j + 8*half), col = ml ----
  float* o = out + (((size_t)n * CONV_H + h) * CONV_W + w0) * CONV_OC;
#pragma unroll
  for (int t = 0; t < MTILES; ++t) {
#pragma unroll
    for (int j = 0; j < 8; ++j) {
      o[(size_t)(16 * t + j + 8 * half) * CONV_OC + ml] = acc[t][j];
    }
  }
}

extern "C" void kernel_launch(void* const* d_in, const int* in_sizes, int n_in,
                              void* d_out, int out_size, void* d_ws, size_t ws_size,
                              hipStream_t stream) {
  const float* x = (const float*)d_in[0];   // (32,128,128,8) NHWC f32
  const float* w = (const float*)d_in[1];   // (3,3,8,16) HWIO f32
  float* out = (float*)d_out;               // (32,128,128,16) NHWC f32

  const int jobs   = 32 * 128 * (128 / 64); // 8192 waves (64-wide M-tiles)
  const int blocks = jobs / 8;              // 8 waves (256 thr) per block
  conv2d_nhwc_wmma_f32<<<blocks, 256, 0, stream>>>(x, w, out);
}